// eGCN_51788715655810
// MI455X (gfx1250) — compile-verified
//
#include <hip/hip_runtime.h>

typedef __attribute__((ext_vector_type(2))) float v2f;
typedef __attribute__((ext_vector_type(8))) float v8f;

#define D_HID 16

// ---------------- degree / normalization ----------------

__global__ void k_init_deg(float* __restrict__ deg, int n) {
  int i = blockIdx.x * blockDim.x + threadIdx.x;
  if (i < n) deg[i] = 1.0f;  // self loop contributes 1 to every node's degree
}

__global__ void k_accum_deg(const long long* __restrict__ dst, float* __restrict__ deg, int E) {
  int e = blockIdx.x * blockDim.x + threadIdx.x;
  if (e < E) atomicAdd(&deg[(int)dst[e]], 1.0f);
}

__global__ void k_make_dinv(float* __restrict__ deg, int n) {
  int i = blockIdx.x * blockDim.x + threadIdx.x;
  if (i < n) {
    float d = deg[i];
    deg[i] = d > 0.0f ? rsqrtf(d) : 0.0f;  // deg >= 1 always, but match reference
  }
}

// ---------------- edge precompute: int64 -> int32 + fused coefficient -------

__global__ void k_prep_edges(const long long* __restrict__ src64,
                             const long long* __restrict__ dst64,
                             const float* __restrict__ dinv,
                             int* __restrict__ src32, int* __restrict__ dst32,
                             float* __restrict__ coef, int E) {
  int e = blockIdx.x * blockDim.x + threadIdx.x;
  if (e >= E) return;
  int s = (int)src64[e];
  int d = (int)dst64[e];
  src32[e] = s;
  dst32[e] = d;
  coef[e] = dinv[s] * dinv[d];
}

// ---------------- WMMA GEMM + fused self-loop/bias epilogue ----------------
// H[n,16] = A[n,K] @ W[K,16];  AGG[n,16] = H * dinv^2 + bias
// One wave computes one 16x16 output tile via K/4 accumulating
// V_WMMA_F32_16X16X4_F32 ops. W is staged in LDS (cooperative copy).
// A fragment (32-bit 16x4): lane L holds row M=L%16; VGPR0 holds
// K = (L<16 ? 0 : 2), VGPR1 holds K+1 -> contiguous float2 per lane.
// Tail rows are CLAMPED (not predicated): only D rows >= n are corrupted and
// those are never stored, so EXEC stays all-1s with no branches in the loop.
// In-place A==AGG is safe: a wave reads only its own 16 rows, then writes them.

template <int K>
__global__ void k_gemm_wmma(const float* __restrict__ A,
                            const float* __restrict__ W,
                            const float* __restrict__ dinv,
                            const float* __restrict__ bias,
                            float* __restrict__ H,
                            float* __restrict__ AGG, int n) {
  __shared__ float sW[K * D_HID];
  for (int t = threadIdx.x; t < K * D_HID; t += blockDim.x) sW[t] = W[t];
  __syncthreads();

  const int lane = threadIdx.x & 31;
  const int wave = threadIdx.x >> 5;
  const int tile = blockIdx.x * (blockDim.x >> 5) + wave;
  const int row0 = tile * 16;
  if (row0 >= n) return;  // wave-uniform exit (after barrier)

  const int m  = lane & 15;
  const int kh = (lane >> 4) << 1;  // 0 or 2
  int row = row0 + m;
  if (row >= n) row = n - 1;  // clamp: unconditional loads, garbage rows unsaved

  const float bcol = bias[m];

  v8f acc = {};
#pragma unroll 4
  for (int kb = 0; kb < K; kb += 4) {
    v2f a;
    const float* ap = A + (size_t)row * K + (kb + kh);
    a.x = ap[0];
    a.y = ap[1];
    v2f b;
    const float* wp = sW + (kb + kh) * D_HID + m;
    b.x = wp[0];
    b.y = wp[D_HID];
    acc = __builtin_amdgcn_wmma_f32_16x16x4_f32(
        /*neg_a=*/false, a, /*neg_b=*/false, b,
        /*c_mod=*/(short)0, acc, /*reuse_a=*/false, /*reuse_b=*/false);
  }

  const int mbase = row0 + ((lane >> 4) << 3);  // +0 or +8
#pragma unroll
  for (int r = 0; r < 8; ++r) {
    int rr = mbase + r;
    if (rr < n) {
      float v = acc[r];
      H[(size_t)rr * D_HID + m] = v;
      float di = dinv[rr];
      AGG[(size_t)rr * D_HID + m] = v * (di * di) + bcol;
    }
  }
}

// ---------------- edge scatter: agg[dst] += h[src] * coef ------------------
// One thread per (edge, channel): 16 consecutive lanes share one edge, so
// h[src*16+c] reads and agg[dst*16+c] atomics are coalesced 64B groups, and
// the per-edge scalars are single broadcast fetches. agg (6.4MB) and the
// edge stream live in the 192MB L2, so this phase is L2-atomic bound.

__global__ void k_scatter_pre(const int* __restrict__ src, const int* __restrict__ dst,
                              const float* __restrict__ coef,
                              const float* __restrict__ h, float* __restrict__ agg, int E) {
  long long idx = (long long)blockIdx.x * blockDim.x + threadIdx.x;
  if (idx >= (long long)E * D_HID) return;
  int e = (int)(idx >> 4);
  int c = (int)(idx & 15);
  atomicAdd(&agg[(size_t)dst[e] * D_HID + c], h[(size_t)src[e] * D_HID + c] * coef[e]);
}

// Fallback (small workspace): read int64 edges + dinv directly.
__global__ void k_scatter(const long long* __restrict__ src, const long long* __restrict__ dst,
                          const float* __restrict__ h, const float* __restrict__ dinv,
                          float* __restrict__ agg, int E) {
  long long idx = (long long)blockIdx.x * blockDim.x + threadIdx.x;
  if (idx >= (long long)E * D_HID) return;
  int e = (int)(idx >> 4);
  int c = (int)(idx & 15);
  int s = (int)src[e];
  int d = (int)dst[e];
  float cf = dinv[s] * dinv[d];
  atomicAdd(&agg[(size_t)d * D_HID + c], h[(size_t)s * D_HID + c] * cf);
}

// ---------------- elementwise tanh ----------------

__global__ void k_tanh(const float* __restrict__ in, float* __restrict__ out, int n) {
  int i = blockIdx.x * blockDim.x + threadIdx.x;
  if (i < n) out[i] = tanhf(in[i]);
}

// ---------------- launch ----------------

extern "C" void kernel_launch(void* const* d_in, const int* in_sizes, int n_in,
                              void* d_out, int out_size, void* d_ws, size_t ws_size,
                              hipStream_t stream) {
  const float* x      = (const float*)d_in[0];
  const long long* ei = (const long long*)d_in[1];  // int64 edge_index [2, E]
  const float* W1     = (const float*)d_in[2];
  const float* b1     = (const float*)d_in[3];
  const float* W2     = (const float*)d_in[4];
  const float* b2     = (const float*)d_in[5];

  const int N = in_sizes[0] / 128;  // 100000
  const int E = in_sizes[1] / 2;    // 3200000
  const long long* src = ei;
  const long long* dst = ei + E;

  float* dinv = (float*)d_ws;                  // [N]
  float* bufA = dinv + N;                      // [N*16]
  float* bufB = bufA + (size_t)N * D_HID;      // [N*16]
  int*   src32 = (int*)(bufB + (size_t)N * D_HID);  // [E]
  int*   dst32 = src32 + E;                         // [E]
  float* coef  = (float*)(dst32 + E);               // [E]
  float* out  = (float*)d_out;

  const size_t need_full = ((size_t)N * (1 + 2 * D_HID) + (size_t)E * 3) * 4;
  const bool pre = ws_size >= need_full;

  const int T = 256;  // 8 wave32 per block
  const int gN  = (N + T - 1) / T;
  const int gE1 = (E + T - 1) / T;
  const int gNC = (N * D_HID + T - 1) / T;
  const long long totE = (long long)E * D_HID;
  const int gEC = (int)((totE + T - 1) / T);
  const int rowsPerBlock = (T / 32) * 16;  // 128 rows per block
  const int gG = (N + rowsPerBlock - 1) / rowsPerBlock;

  // dinv = rsqrt(1 + in_degree)
  k_init_deg<<<gN, T, 0, stream>>>(dinv, N);
  k_accum_deg<<<gE1, T, 0, stream>>>(dst, dinv, E);
  k_make_dinv<<<gN, T, 0, stream>>>(dinv, N);
  if (pre) k_prep_edges<<<gE1, T, 0, stream>>>(src, dst, dinv, src32, dst32, coef, E);

  // layer 1: h1 = x@W1 -> bufA ; agg1 = h1*dinv^2 + b1 -> bufB (fused)
  k_gemm_wmma<128><<<gG, T, 0, stream>>>(x, W1, dinv, b1, bufA, bufB, N);
  if (pre) k_scatter_pre<<<gEC, T, 0, stream>>>(src32, dst32, coef, bufA, bufB, E);
  else     k_scatter<<<gEC, T, 0, stream>>>(src, dst, bufA, dinv, bufB, E);
  k_tanh<<<gNC, T, 0, stream>>>(bufB, bufA, N * D_HID);  // t1 -> bufA

  // layer 2: h2 = t1@W2 -> bufB ; agg2 -> bufA (in-place over t1: safe per-wave)
  k_gemm_wmma<16><<<gG, T, 0, stream>>>(bufA, W2, dinv, b2, bufB, bufA, N);
  if (pre) k_scatter_pre<<<gEC, T, 0, stream>>>(src32, dst32, coef, bufB, bufA, E);
  else     k_scatter<<<gEC, T, 0, stream>>>(src, dst, bufB, dinv, bufA, E);
  k_tanh<<<gNC, T, 0, stream>>>(bufA, out, N * D_HID);
}